// PatchPlaneApproxLoss_40931038331572
// MI455X (gfx1250) — compile-verified
//
#include <hip/hip_runtime.h>
#include <hip/hip_bf16.h>

typedef __attribute__((ext_vector_type(2))) float v2f;
typedef __attribute__((ext_vector_type(8))) float v8f;

#define IMG_H 479
#define IMG_W 639
#define IMG_B 8
#define TILES_X 40
#define TILES_Y 30
#define NBLOCKS (IMG_B * TILES_X * TILES_Y)   // 9600
#define EPSF 1e-6f
#define INV_FX (1.0f / 518.857f)
#define INV_FY (1.0f / 519.469f)

// D = A(16x4) x B(4x16) + C, f32 WMMA (wave32).
static __device__ inline v8f wmma4(v2f a, v2f b, v8f c) {
    return __builtin_amdgcn_wmma_f32_16x16x4_f32(false, a, false, b, (short)0, c,
                                                 false, false);
}

// Banded-ones matrix: S[m,k] = 1 iff m <= k <= m+7 (8-wide box window).
static __device__ inline float band(int m, int k) {
    return (k >= m && k <= m + 7) ? 1.0f : 0.0f;
}

// Moment for channel group g (0: x*{x,y,z}, 1: {yy,yz,zz}, 2: {x,y,z}), ch 0..2
static __device__ inline float gmom(int g, int ch, float x, float y, float z) {
    if (g == 0) return x * (ch == 0 ? x : (ch == 1 ? y : z));
    if (g == 1) return (ch == 0) ? y * y : (ch == 1 ? y * z : z * z);
    return (ch == 0) ? x : (ch == 1 ? y : z);
}

// One wave (32 threads) per 16x16 output tile.
__global__ __launch_bounds__(32) void ppal_tile_kernel(
    const float* __restrict__ pred, const float* __restrict__ gt,
    float* __restrict__ partial) {
    __shared__ float lds_d[32][24];       // depth tile + halo, zero padded
    __shared__ float lds_hs[3][16][24];   // pass-1 hs^T[ch][c_out][r]

    const int lane = threadIdx.x;   // 0..31
    const int m  = lane & 15;
    const int hi = lane >> 4;

    const int blk = blockIdx.x;
    const int b   = blk / (TILES_X * TILES_Y);
    const int t   = blk % (TILES_X * TILES_Y);
    const int ti  = t / TILES_X;
    const int tj  = t % TILES_X;
    const int i0  = ti * 16;
    const int j0  = tj * 16;

    // Zero the never-written column 23 of lds_hs once (uninitialized LDS could
    // contain NaN patterns; 0*NaN would poison the WMMA accumulators).
    for (int idx = lane; idx < 3 * 16; idx += 32) lds_hs[idx / 16][idx % 16][23] = 0.0f;

    // Per-lane constant banded-ones A operands, shared by both passes.
    v2f Aband[6];
#pragma unroll
    for (int kc = 0; kc < 6; ++kc) {
        const int k0 = 4 * kc + 2 * hi;
        Aband[kc] = {band(m, k0), band(m, k0 + 1)};
    }

    float npx[8], npy[8], npz[8];
    float acc = 0.0f;
    const v8f vzero = {0.f, 0.f, 0.f, 0.f, 0.f, 0.f, 0.f, 0.f};

    for (int img = 0; img < 2; ++img) {
        const float* depth = (img == 0 ? pred : gt) + (size_t)b * (IMG_H * IMG_W);

        __syncthreads();
        // Fill the whole 32x24 tile: clamped unconditional load + select (no
        // EXEC-masked VMEM), zeros everywhere outside the 23x23 halo window.
        for (int idx = lane; idx < 32 * 24; idx += 32) {
            const int r  = idx / 24;
            const int e  = idx % 24;
            const int gi = i0 - 4 + r;
            const int gj = j0 - 4 + e;
            const bool inb = (r < 23) && (e < 23) && (gi >= 0) && (gi < IMG_H) &&
                             (gj >= 0) && (gj < IMG_W);
            const int gic = min(max(gi, 0), IMG_H - 1);
            const int gjc = min(max(gj, 0), IMG_W - 1);
            const float dv = depth[gic * IMG_W + gjc];
            lds_d[r][e] = inb ? dv : 0.0f;
        }
        __syncthreads();

        v8f S[9];
#pragma unroll
        for (int c = 0; c < 9; ++c) S[c] = vzero;

#pragma unroll
        for (int g = 0; g < 3; ++g) {
            // ---- Pass 1 (horizontal): hs^T = Sh(16x24) x mom^T(24x23) ----
            v8f D0[3] = {vzero, vzero, vzero};
            v8f D1[3] = {vzero, vzero, vzero};
#pragma unroll
            for (int kc = 0; kc < 6; ++kc) {
                const int k0 = 4 * kc + 2 * hi;
                const v2f Av = Aband[kc];
                // Depth once per chunk (pairs merge into ds_load_b64).
                const float d00 = lds_d[m][k0],      d01 = lds_d[m][k0 + 1];
                const float d10 = lds_d[m + 16][k0], d11 = lds_d[m + 16][k0 + 1];
                const float u0 = (float)(j0 - 4 + k0) - 319.5f;
                const float u1 = u0 + 1.0f;
                const float va = (float)(i0 - 4 + m) - 239.5f;
                const float vb = va + 16.0f;
                const float x00 = u0 * d00 * INV_FX, y00 = va * d00 * INV_FY;
                const float x01 = u1 * d01 * INV_FX, y01 = va * d01 * INV_FY;
                const float x10 = u0 * d10 * INV_FX, y10 = vb * d10 * INV_FY;
                const float x11 = u1 * d11 * INV_FX, y11 = vb * d11 * INV_FY;
#pragma unroll
                for (int ch = 0; ch < 3; ++ch) {
                    const v2f B0 = {gmom(g, ch, x00, y00, d00),
                                    gmom(g, ch, x01, y01, d01)};
                    D0[ch] = wmma4(Av, B0, D0[ch]);
                    const v2f B1 = {gmom(g, ch, x10, y10, d10),
                                    gmom(g, ch, x11, y11, d11)};
                    D1[ch] = wmma4(Av, B1, D1[ch]);
                }
            }

            __syncthreads();   // prior group's pass-2 reads complete
#pragma unroll
            for (int ch = 0; ch < 3; ++ch)
#pragma unroll
                for (int v = 0; v < 8; ++v) lds_hs[ch][v + 8 * hi][m] = D0[ch][v];
            if (m < 7) {
#pragma unroll
                for (int ch = 0; ch < 3; ++ch)
#pragma unroll
                    for (int v = 0; v < 8; ++v)
                        lds_hs[ch][v + 8 * hi][16 + m] = D1[ch][v];
            }
            __syncthreads();

            // ---- Pass 2 (vertical): out = Sv(16x24) x hs(24x16) ----
#pragma unroll
            for (int kc = 0; kc < 6; ++kc) {
                const int k0 = 4 * kc + 2 * hi;
                const v2f Av = Aband[kc];
#pragma unroll
                for (int ch = 0; ch < 3; ++ch) {
                    const v2f Bv = {lds_hs[ch][m][k0], lds_hs[ch][m][k0 + 1]};
                    S[3 * g + ch] = wmma4(Av, Bv, S[3 * g + ch]);
                }
            }
        }

        // ---- Per-pixel 3x3 solve, normalize, flip, mask (8 pixels / lane) ----
#pragma unroll
        for (int v = 0; v < 8; ++v) {
            const int mrow = v + 8 * hi;
            const int ncol = m;
            const int gi = i0 + mrow;
            const int gj = j0 + ncol;
            const bool valid = (gi < IMG_H) && (gj < IMG_W);

            const float a  = S[0][v] + EPSF;  // xx
            const float d  = S[1][v];         // xy
            const float e  = S[2][v];         // xz
            const float bb = S[3][v] + EPSF;  // yy
            const float f  = S[4][v];         // yz
            const float cc = S[5][v] + EPSF;  // zz
            const float p  = S[6][v];         // xs
            const float q  = S[7][v];         // ys
            const float rr = S[8][v];         // zs

            const float A11 = bb * cc - f * f;
            const float A12 = d * cc - f * e;
            const float A13 = d * f - bb * e;
            const float det = a * A11 - d * A12 + e * A13;

            const float qcr = q * cc - f * rr;
            const float dre = d * rr - q * e;
            const float detx = p * A11 - d * qcr + e * (q * f - bb * rr);
            const float dety = a * qcr - p * A12 + e * dre;
            const float detz = a * (bb * rr - q * f) - d * dre + p * A13;

            const float len2 = detx * detx + dety * dety + detz * detz;
            float inv = rsqrtf(len2);
            if (det < 0.0f) inv = -inv;       // n = sign(det)*detv/||detv||
            float nx = detx * inv, ny = dety * inv, nz = detz * inv;

            const float dc = lds_d[mrow + 4][ncol + 4];
            const float px = ((float)gj - 319.5f) * dc * INV_FX;
            const float py = ((float)gi - 239.5f) * dc * INV_FY;
            const float dot = nx * px + ny * py + nz * dc;
            if (dot > 0.0f) { nx = -nx; ny = -ny; nz = -nz; }
            if (!(dc > 0.0f)) { nx = 0.0f; ny = 0.0f; nz = 0.0f; }

            if (img == 0) {
                npx[v] = nx; npy[v] = ny; npz[v] = nz;
            } else if (valid) {
                acc += fabsf(nx - npx[v]) + fabsf(ny - npy[v]) + fabsf(nz - npz[v]);
            }
        }
    }

    // Wave32 reduction; lane 0 writes block partial (deterministic).
#pragma unroll
    for (int off = 16; off > 0; off >>= 1) acc += __shfl_xor(acc, off, 32);
    if (lane == 0) partial[blockIdx.x] = acc;
}

__global__ __launch_bounds__(256) void ppal_reduce_kernel(
    const float* __restrict__ partial, float* __restrict__ out, int n) {
    __shared__ float sm[256];
    float s = 0.0f;
    for (int i = threadIdx.x; i < n; i += 256) s += partial[i];
    sm[threadIdx.x] = s;
    __syncthreads();
    for (int w = 128; w > 0; w >>= 1) {
        if (threadIdx.x < w) sm[threadIdx.x] += sm[threadIdx.x + w];
        __syncthreads();
    }
    if (threadIdx.x == 0) {
        const float scale = 1.0f / (8.0f * 3.0f * 479.0f * 639.0f);
        out[0] = sm[0] * scale;
    }
}

extern "C" void kernel_launch(void* const* d_in, const int* in_sizes, int n_in,
                              void* d_out, int out_size, void* d_ws, size_t ws_size,
                              hipStream_t stream) {
    const float* pred = (const float*)d_in[0];
    const float* gt   = (const float*)d_in[1];
    float* out        = (float*)d_out;
    float* partial    = (float*)d_ws;   // NBLOCKS floats of scratch

    ppal_tile_kernel<<<NBLOCKS, 32, 0, stream>>>(pred, gt, partial);
    ppal_reduce_kernel<<<1, 256, 0, stream>>>(partial, out, NBLOCKS);
}